// ProbAttention_52415780880839
// MI455X (gfx1250) — compile-verified
//
#include <hip/hip_runtime.h>

// ProbSparse attention for MI455X (gfx1250, wave32).
// Bandwidth-bound: ~600MB HBM traffic -> ~26us floor @23.3TB/s.
// All GEMMs use V_WMMA_F32_16X16X4_F32 (fp32, matches reference numerics).

#define B_ 16
#define H_ 8
#define L_ 4096
#define S_ 4096
#define D_ 64
#define U_ 45
#define UP_ 48
#define BH_ (B_ * H_)
#define NPART_ 16
#define SCALE_ 0.125f  // 1/sqrt(64)

typedef float v2f __attribute__((ext_vector_type(2)));
typedef float v8f __attribute__((ext_vector_type(8)));

__device__ __forceinline__ v8f wmma_f32(v2f a, v2f b, v8f c) {
  // D = A(16x4) * B(4x16) + C(16x16), fp32
  return __builtin_amdgcn_wmma_f32_16x16x4_f32(false, a, false, b, (short)0, c,
                                               false, false);
}

// ---------------------------------------------------------------------------
// Kernel 1: M[bh][l] = max_k(Q.Ks^T) - sum_k(Q.Ks^T)/S over 45 sampled keys.
// grid (BH, 8), block 256 (8 waves); each wave does 4 tiles of 16 queries.
// ---------------------------------------------------------------------------
__global__ __launch_bounds__(256) void k1_sample_scores(
    const float* __restrict__ Qg, const float* __restrict__ Kg,
    const int* __restrict__ sidx, float* __restrict__ Mout) {
  const int bh = blockIdx.x;
  const int chunk = blockIdx.y;  // 0..7
  __shared__ float Ks[UP_][D_ + 2];  // padded stride to spread LDS banks

  const float* Kbh = Kg + (size_t)bh * S_ * D_;
  const float* Qbh = Qg + (size_t)bh * L_ * D_;

  for (int i = threadIdx.x; i < UP_ * D_; i += 256) {
    const int u = i >> 6, d = i & 63;
    float v = 0.0f;
    if (u < U_) v = Kbh[(size_t)sidx[u] * D_ + d];
    Ks[u][d] = v;
  }
  __syncthreads();

  const int wave = threadIdx.x >> 5;
  const int lane = threadIdx.x & 31;
  const int half = lane >> 4;
  const int r = lane & 15;

  for (int i = 0; i < 4; ++i) {
    const int tile = chunk * 32 + wave * 4 + i;
    const int l0 = tile * 16;
    v8f acc0 = {}, acc1 = {}, acc2 = {};
#pragma unroll
    for (int s = 0; s < 16; ++s) {
      const int d0 = 4 * s + 2 * half;
      v2f a = *(const v2f*)(Qbh + (size_t)(l0 + r) * D_ + d0);  // A: Q rows
      v2f b0 = *(const v2f*)(&Ks[r][d0]);        // B^T: sampled-key rows
      v2f b1 = *(const v2f*)(&Ks[16 + r][d0]);
      v2f b2 = *(const v2f*)(&Ks[32 + r][d0]);
      acc0 = wmma_f32(a, b0, acc0);
      acc1 = wmma_f32(a, b1, acc1);
      acc2 = wmma_f32(a, b2, acc2);
    }
    // C layout: lane(h*16+n) vgpr j -> C[l = j+8h][key = 16t+n]
    float pm[8], ps[8];
#pragma unroll
    for (int j = 0; j < 8; ++j) {
      float mx = fmaxf(acc0[j], acc1[j]);
      float sm = acc0[j] + acc1[j];
      if (32 + r < U_) { mx = fmaxf(mx, acc2[j]); sm += acc2[j]; }
      pm[j] = mx;
      ps[j] = sm;
    }
#pragma unroll
    for (int msk = 1; msk < 16; msk <<= 1) {
#pragma unroll
      for (int j = 0; j < 8; ++j) {
        pm[j] = fmaxf(pm[j], __shfl_xor(pm[j], msk, 32));
        ps[j] += __shfl_xor(ps[j], msk, 32);
      }
    }
    if (r == 0) {
#pragma unroll
      for (int j = 0; j < 8; ++j)
        Mout[(size_t)bh * L_ + l0 + j + 8 * half] =
            pm[j] - ps[j] * (1.0f / (float)S_);
    }
  }
}

// ---------------------------------------------------------------------------
// Kernel 2: top-45 indices of M per (b,h). Tie-break = lowest index.
// ---------------------------------------------------------------------------
__global__ __launch_bounds__(256) void k2_topk(const float* __restrict__ Mv,
                                               int* __restrict__ Mtop) {
  const int bh = blockIdx.x;
  __shared__ float sm[L_];
  __shared__ float rv[256];
  __shared__ int ri[256];
  for (int i = threadIdx.x; i < L_; i += 256) sm[i] = Mv[(size_t)bh * L_ + i];
  __syncthreads();
  for (int it = 0; it < U_; ++it) {
    float best = -__builtin_huge_valf();
    int bi = L_;
    for (int i = threadIdx.x; i < L_; i += 256) {
      const float v = sm[i];
      if (v > best || (v == best && i < bi)) { best = v; bi = i; }
    }
    rv[threadIdx.x] = best;
    ri[threadIdx.x] = bi;
    __syncthreads();
    for (int s = 128; s > 0; s >>= 1) {
      if (threadIdx.x < s) {
        const float v2 = rv[threadIdx.x + s];
        const int i2 = ri[threadIdx.x + s];
        if (v2 > rv[threadIdx.x] ||
            (v2 == rv[threadIdx.x] && i2 < ri[threadIdx.x])) {
          rv[threadIdx.x] = v2;
          ri[threadIdx.x] = i2;
        }
      }
      __syncthreads();
    }
    if (threadIdx.x == 0) {
      const int w = ri[0];
      Mtop[bh * UP_ + it] = w;
      sm[w] = -__builtin_huge_valf();
    }
    __syncthreads();
  }
  if (threadIdx.x < UP_ - U_) Mtop[bh * UP_ + U_ + threadIdx.x] = 0;
}

// ---------------------------------------------------------------------------
// Kernel 3: flash attention for the 45 (padded 48) selected queries.
// grid (BH, 2), block 256 (8 waves): each wave = one split of 256 keys,
// independent online-softmax partial; no cross-wave merge needed.
// ---------------------------------------------------------------------------
__global__ __launch_bounds__(256) void k3_flash(
    const float* __restrict__ Qg, const float* __restrict__ Kg,
    const float* __restrict__ Vg, const int* __restrict__ Mtop,
    float* __restrict__ pm, float* __restrict__ pl, float* __restrict__ pacc,
    float* __restrict__ pvs) {
  const int bh = blockIdx.x;
  const int wave = threadIdx.x >> 5;
  const int lane = threadIdx.x & 31;
  const int half = lane >> 4;
  const int r = lane & 15;
  const int part = blockIdx.y * 8 + wave;  // 0..15

  __shared__ float Qr[UP_][D_ + 2];   // gathered top queries
  __shared__ float Pb[8][UP_][18];    // per-wave prob tile (C->A relayout)

  const int* mt = Mtop + bh * UP_;
  for (int i = threadIdx.x; i < UP_ * D_; i += 256) {
    const int u = i >> 6, d = i & 63;
    float v = 0.0f;
    if (u < U_) v = Qg[((size_t)bh * L_ + mt[u]) * D_ + d];
    Qr[u][d] = v;
  }
  __syncthreads();

  float m[3][8], l[3][8];
  v8f acc[3][4];
  float vsm[4] = {0.f, 0.f, 0.f, 0.f};
#pragma unroll
  for (int t = 0; t < 3; ++t) {
#pragma unroll
    for (int j = 0; j < 8; ++j) {
      m[t][j] = -__builtin_huge_valf();
      l[t][j] = 0.0f;
    }
#pragma unroll
    for (int dt = 0; dt < 4; ++dt) acc[t][dt] = (v8f){};
  }

  for (int c = 0; c < 16; ++c) {
    const int kb = part * 256 + c * 16;
    const float* Kb = Kg + ((size_t)bh * S_ + kb) * D_;
    const float* Vb = Vg + ((size_t)bh * S_ + kb) * D_;

    // scores tile: C[u][key], 48x16, K-dim = 64
    v8f scs[3] = {(v8f){}, (v8f){}, (v8f){}};
#pragma unroll
    for (int s = 0; s < 16; ++s) {
      const int d0 = 4 * s + 2 * half;
      v2f b = *(const v2f*)(Kb + (size_t)r * D_ + d0);  // B^T: key rows
      v2f a0 = *(const v2f*)(&Qr[r][d0]);               // A: query rows
      v2f a1 = *(const v2f*)(&Qr[16 + r][d0]);
      v2f a2 = *(const v2f*)(&Qr[32 + r][d0]);
      scs[0] = wmma_f32(a0, b, scs[0]);
      scs[1] = wmma_f32(a1, b, scs[1]);
      scs[2] = wmma_f32(a2, b, scs[2]);
    }

    // online softmax; C layout: lane(h*16+n) vgpr j -> u=16t+j+8h, key=kb+n
#pragma unroll
    for (int t = 0; t < 3; ++t) {
      float cm[8], p[8];
#pragma unroll
      for (int j = 0; j < 8; ++j) {
        scs[t][j] *= SCALE_;
        cm[j] = scs[t][j];
      }
#pragma unroll
      for (int msk = 1; msk < 16; msk <<= 1)
#pragma unroll
        for (int j = 0; j < 8; ++j)
          cm[j] = fmaxf(cm[j], __shfl_xor(cm[j], msk, 32));
#pragma unroll
      for (int j = 0; j < 8; ++j) {
        const float mn = fmaxf(m[t][j], cm[j]);
        const float corr = expf(m[t][j] - mn);
        m[t][j] = mn;
        l[t][j] *= corr;
#pragma unroll
        for (int dt = 0; dt < 4; ++dt) acc[t][dt][j] *= corr;
        const float pv = expf(scs[t][j] - mn);
        Pb[wave][16 * t + j + 8 * half][r] = pv;  // [u][key] for A-frags
        p[j] = pv;
      }
#pragma unroll
      for (int msk = 1; msk < 16; msk <<= 1)
#pragma unroll
        for (int j = 0; j < 8; ++j) p[j] += __shfl_xor(p[j], msk, 32);
#pragma unroll
      for (int j = 0; j < 8; ++j) l[t][j] += p[j];
    }

    // PV: attn(48x16) @ V(16x64); accumulator u-mapping matches m/l state
#pragma unroll
    for (int s = 0; s < 4; ++s) {
      const int k0 = 4 * s + 2 * half;
      v2f af0 = *(const v2f*)(&Pb[wave][r][k0]);
      v2f af1 = *(const v2f*)(&Pb[wave][16 + r][k0]);
      v2f af2 = *(const v2f*)(&Pb[wave][32 + r][k0]);
#pragma unroll
      for (int dt = 0; dt < 4; ++dt) {
        v2f bf;
        bf.x = Vb[(size_t)k0 * D_ + 16 * dt + r];
        bf.y = Vb[(size_t)(k0 + 1) * D_ + 16 * dt + r];
        vsm[dt] += bf.x + bf.y;  // free V row-sum partial
        acc[0][dt] = wmma_f32(af0, bf, acc[0][dt]);
        acc[1][dt] = wmma_f32(af1, bf, acc[1][dt]);
        acc[2][dt] = wmma_f32(af2, bf, acc[2][dt]);
      }
    }
  }

  // write deterministic per-wave partials
#pragma unroll
  for (int dt = 0; dt < 4; ++dt) vsm[dt] += __shfl_xor(vsm[dt], 16, 32);
  const size_t pb = (size_t)bh * NPART_ + part;
  if (half == 0) {
#pragma unroll
    for (int dt = 0; dt < 4; ++dt) pvs[pb * D_ + 16 * dt + r] = vsm[dt];
  }
  if (r == 0) {
#pragma unroll
    for (int t = 0; t < 3; ++t)
#pragma unroll
      for (int j = 0; j < 8; ++j) {
        const int u = 16 * t + j + 8 * half;
        pm[pb * UP_ + u] = m[t][j];
        pl[pb * UP_ + u] = l[t][j];
      }
  }
#pragma unroll
  for (int t = 0; t < 3; ++t)
#pragma unroll
    for (int dt = 0; dt < 4; ++dt)
#pragma unroll
      for (int j = 0; j < 8; ++j) {
        const int u = 16 * t + j + 8 * half;
        pacc[(pb * UP_ + u) * D_ + 16 * dt + r] = acc[t][dt][j];
      }
}

// ---------------------------------------------------------------------------
// Kernel 4: broadcast V_sum to the whole output. grid (BH, 16), 256 rows/blk.
// ---------------------------------------------------------------------------
__global__ __launch_bounds__(256) void k4_fill(const float* __restrict__ pvs,
                                               float* __restrict__ out) {
  const int bh = blockIdx.x;
  const int c = blockIdx.y;
  __shared__ float vsum[D_];
  if (threadIdx.x < D_) {
    float s = 0.f;
    for (int p = 0; p < NPART_; ++p)
      s += pvs[((size_t)bh * NPART_ + p) * D_ + threadIdx.x];
    vsum[threadIdx.x] = s;
  }
  __syncthreads();
  float4* ob = (float4*)(out + ((size_t)bh * L_ + (size_t)c * 256) * D_);
  for (int idx = threadIdx.x; idx < 256 * 16; idx += 256) {
    const int c4 = idx & 15;
    ob[idx] = *(const float4*)&vsum[c4 * 4];
  }
}

// ---------------------------------------------------------------------------
// Kernel 5: merge the 16 softmax partials (fixed order -> deterministic),
// scatter the 45 context rows over the broadcast fill.
// ---------------------------------------------------------------------------
__global__ __launch_bounds__(256) void k5_scatter(
    const float* __restrict__ pm, const float* __restrict__ pl,
    const float* __restrict__ pacc, const int* __restrict__ Mtop,
    float* __restrict__ out) {
  const int bh = blockIdx.x;
  __shared__ float sm_[NPART_][UP_];
  __shared__ float sl_[NPART_][UP_];
  __shared__ float mg[UP_];
  __shared__ float lg[UP_];
  const size_t base = (size_t)bh * NPART_ * UP_;
  for (int i = threadIdx.x; i < NPART_ * UP_; i += 256) {
    (&sm_[0][0])[i] = pm[base + i];
    (&sl_[0][0])[i] = pl[base + i];
  }
  __syncthreads();
  if (threadIdx.x < UP_) {
    const int u = threadIdx.x;
    float mgv = -__builtin_huge_valf();
    for (int p = 0; p < NPART_; ++p) mgv = fmaxf(mgv, sm_[p][u]);
    float lgv = 0.f;
    for (int p = 0; p < NPART_; ++p) lgv += expf(sm_[p][u] - mgv) * sl_[p][u];
    mg[u] = mgv;
    lg[u] = lgv;
  }
  __syncthreads();
  for (int idx = threadIdx.x; idx < U_ * D_; idx += 256) {
    const int u = idx >> 6, d = idx & 63;
    float s = 0.f;
    for (int p = 0; p < NPART_; ++p)
      s += expf(sm_[p][u] - mg[u]) * pacc[(base + (size_t)p * UP_ + u) * D_ + d];
    out[((size_t)bh * L_ + Mtop[bh * UP_ + u]) * D_ + d] = s / lg[u];
  }
}

// ---------------------------------------------------------------------------
extern "C" void kernel_launch(void* const* d_in, const int* in_sizes, int n_in,
                              void* d_out, int out_size, void* d_ws,
                              size_t ws_size, hipStream_t stream) {
  const float* Q = (const float*)d_in[0];
  const float* K = (const float*)d_in[1];
  const float* V = (const float*)d_in[2];
  const int* sidx = (const int*)d_in[3];
  float* out = (float*)d_out;
  float* ws = (float*)d_ws;

  // workspace layout (floats), total ~28.6 MB
  float* Mv = ws;                                      // BH*L
  int* Mtop = (int*)(ws + (size_t)BH_ * L_);           // BH*UP
  float* pm = ws + (size_t)BH_ * L_ + (size_t)BH_ * UP_;
  float* pl = pm + (size_t)BH_ * NPART_ * UP_;
  float* pacc = pl + (size_t)BH_ * NPART_ * UP_;       // BH*NPART*UP*D
  float* pvs = pacc + (size_t)BH_ * NPART_ * UP_ * D_; // BH*NPART*D

  k1_sample_scores<<<dim3(BH_, 8), 256, 0, stream>>>(Q, K, sidx, Mv);
  k2_topk<<<BH_, 256, 0, stream>>>(Mv, Mtop);
  k3_flash<<<dim3(BH_, 2), 256, 0, stream>>>(Q, K, V, Mtop, pm, pl, pacc, pvs);
  k4_fill<<<dim3(BH_, 16), 256, 0, stream>>>(pvs, out);
  k5_scatter<<<BH_, 256, 0, stream>>>(pm, pl, pacc, Mtop, out);
}